// TopKDecoder_25494925869736
// MI455X (gfx1250) — compile-verified
//
#include <hip/hip_runtime.h>
#include <math.h>

// Problem dims (fixed by reference)
#define B_   8
#define K_   8
#define T_   16
#define V_   32000
#define H_   512
#define S_   50
#define N_   64          // B*K
#define G3H  (3*H_)      // 1536
#define NEGV -1e9f
#define SOS_ 1
#define EOS_ 2
#define KC_  512         // K-chunk staged in LDS (64 rows x 512 = 128 KB)

typedef __attribute__((ext_vector_type(2))) float v2f;
typedef __attribute__((ext_vector_type(8))) float v8f;

// ---------------------------------------------------------------------------
// K0: init hidden / scores / tokens
// ---------------------------------------------------------------------------
__global__ void k_init(const float* __restrict__ dec_hidden,
                       float* __restrict__ h, float* __restrict__ ss,
                       int* __restrict__ tok) {
  int n = blockIdx.x;            // 0..63
  int b = n / K_;
  for (int i = threadIdx.x; i < H_; i += blockDim.x)
    h[n * H_ + i] = dec_hidden[b * H_ + i];
  if (threadIdx.x == 0) {
    ss[n]  = (n % K_ == 0) ? 0.f : NEGV;
    tok[n] = SOS_;
  }
}

// ---------------------------------------------------------------------------
// K1a: attention + context + embedding gather -> X = [emb, ctx] (64 x 1024)
// One beam row per block (256 threads = 8 waves).
// ---------------------------------------------------------------------------
__global__ void k_attn_ctx(const float* __restrict__ h,
                           const int*   __restrict__ tok,
                           const float* __restrict__ emb_tab,
                           const float* __restrict__ q_enc,
                           const int*   __restrict__ q_len,
                           float* __restrict__ X) {   // (N, 2H) row-major
  __shared__ float sh_h[H_];
  __shared__ float sh_att[64];

  int n = blockIdx.x;
  int b = n / K_;
  int tid = threadIdx.x;
  int lane = tid & 31, wave = tid >> 5;
  const float* q = q_enc + (size_t)b * S_ * H_;
  int len = q_len[b];

  for (int i = tid; i < H_; i += blockDim.x) sh_h[i] = h[n * H_ + i];
  __syncthreads();

  // attention scores: wave w handles s = w, w+8, ...
  for (int s = wave; s < S_; s += 8) {
    float p = 0.f;
    for (int i = lane; i < H_; i += 32) p += sh_h[i] * q[s * H_ + i];
    for (int off = 16; off > 0; off >>= 1) p += __shfl_xor(p, off, 32);
    if (lane == 0) sh_att[s] = (s < len) ? p : NEGV;
  }
  __syncthreads();

  // masked softmax over S (tiny; single thread)
  if (tid == 0) {
    float m = -INFINITY;
    for (int s = 0; s < S_; ++s) m = fmaxf(m, sh_att[s]);
    float sum = 0.f;
    for (int s = 0; s < S_; ++s) { float e = expf(sh_att[s] - m); sh_att[s] = e; sum += e; }
    float inv = 1.f / sum;
    for (int s = 0; s < S_; ++s) sh_att[s] *= inv;
  }
  __syncthreads();

  // context + embedding -> X row
  int tk = tok[n];
  for (int i = tid; i < H_; i += blockDim.x) {
    float c = 0.f;
    for (int s = 0; s < S_; ++s) c += sh_att[s] * q[s * H_ + i];
    X[(size_t)n * (2 * H_) + H_ + i] = c;                       // ctx half
    X[(size_t)n * (2 * H_) + i] = emb_tab[(size_t)tk * H_ + i]; // emb half
  }
}

// ---------------------------------------------------------------------------
// Generic batched WMMA GEMM: C(64 x P) = A(64 x Kd) @ W(Kd x P) + bias
// Block = 256 threads (8 waves), covers ALL 64 rows x 128 cols.
// Each wave: one 16-col tile, 4 row-tiles (4 accumulators) -> each B fragment
// feeds 4 v_wmma_f32_16x16x4_f32, and W is streamed exactly once per GEMM.
// A is staged in 128 KB dynamic LDS in K-chunks of 512.
// W addressing is strength-reduced to running pointers (no per-iter muls).
// Fragment layouts per ISA 7.12.2.
// ---------------------------------------------------------------------------
__global__ void k_gemm_wmma(const float* __restrict__ A,
                            const float* __restrict__ W,
                            const float* __restrict__ bias,
                            float* __restrict__ C,
                            int Kd, int P) {
  extern __shared__ float shA[];          // 64 x KC_ = 128 KB
  int tid = threadIdx.x;
  int lane = tid & 31, wave = tid >> 5;   // 8 waves
  int col0 = blockIdx.x * 128 + wave * 16;
  int mrow = lane & 15;                   // A row within tile / B,C column lane
  int hi   = lane >> 4;                   // K-pair select
  int ncol = col0 + mrow;

  const size_t stepP = (size_t)4 * P;     // W advance per K-step (floats)
  v8f acc[4] = {};                        // row tiles m0 = 0,16,32,48

  for (int kc = 0; kc < Kd; kc += KC_) {
    __syncthreads();                      // previous chunk fully consumed
    // cooperative load A[:, kc:kc+512] -> shA (vectorized 128-bit)
    {
      const float4* src4 = (const float4*)A;
      float4* dst4 = (float4*)shA;
      int kd4 = Kd >> 2, kc4 = kc >> 2;
      for (int i4 = tid; i4 < (N_ * KC_) / 4; i4 += 256) {
        int row = i4 >> 7, c4 = i4 & 127; // KC_/4 = 128
        dst4[i4] = src4[row * kd4 + kc4 + c4];
      }
    }
    __syncthreads();

    // running W pointer: row (kc + 2*hi), column ncol
    const float* wptr = W + (size_t)(kc + 2 * hi) * P + ncol;
    int aidx = mrow * KC_ + 2 * hi;       // running LDS index

    for (int k0 = 0; k0 < KC_; k0 += 4) {
      v2f bm;
      bm.x = wptr[0];                     // row k, this column
      bm.y = wptr[(size_t)P];             // row k+1
      if ((k0 & 63) == 0)                 // uniform scalar cond (EXEC stays ~0)
        __builtin_prefetch(wptr + (size_t)64 * P, 0, 3);  // near-cache prefetch
      #pragma unroll
      for (int r = 0; r < 4; ++r) {
        v2f a;
        a.x = shA[aidx + r * (16 * KC_)];
        a.y = shA[aidx + r * (16 * KC_) + 1];
        acc[r] = __builtin_amdgcn_wmma_f32_16x16x4_f32(
            false, a, false, bm, (short)0, acc[r], false, false);
      }
      wptr += stepP;
      aidx += 4;
    }
  }

  float bv = bias[ncol];
  #pragma unroll
  for (int r = 0; r < 4; ++r) {
    #pragma unroll
    for (int j = 0; j < 8; ++j) {
      int row = r * 16 + j + hi * 8;      // C layout: vgpr j -> M=j / M=j+8
      C[(size_t)row * P + ncol] = acc[r][j] + bv;
    }
  }
}

// ---------------------------------------------------------------------------
// K1c: GRU elementwise combine: h_new from gi, gh, h
// ---------------------------------------------------------------------------
__global__ void k_gru_elem(const float* __restrict__ gi,
                           const float* __restrict__ gh,
                           const float* __restrict__ h,
                           float* __restrict__ h_new) {
  int n = blockIdx.x;
  const float* gin = gi + (size_t)n * G3H;
  const float* ghn = gh + (size_t)n * G3H;
  for (int j = threadIdx.x; j < H_; j += blockDim.x) {
    float r  = 1.f / (1.f + expf(-(gin[j]        + ghn[j])));
    float z  = 1.f / (1.f + expf(-(gin[H_ + j]   + ghn[H_ + j])));
    float nn = tanhf(gin[2 * H_ + j] + r * ghn[2 * H_ + j]);
    h_new[n * H_ + j] = (1.f - z) * nn + z * h[n * H_ + j];
  }
}

// ---------------------------------------------------------------------------
// K3: per-row max + log-sum-exp over V (block per row)
// ---------------------------------------------------------------------------
__global__ void k_row_stats(const float* __restrict__ logits,
                            float* __restrict__ rowmax, float* __restrict__ rowlse) {
  __shared__ float red[256];
  int n = blockIdx.x, tid = threadIdx.x;
  const float* row = logits + (size_t)n * V_;
  float m = -INFINITY;
  for (int v = tid; v < V_; v += 256) m = fmaxf(m, row[v]);
  red[tid] = m; __syncthreads();
  for (int s = 128; s > 0; s >>= 1) { if (tid < s) red[tid] = fmaxf(red[tid], red[tid + s]); __syncthreads(); }
  m = red[0]; __syncthreads();
  float sum = 0.f;
  for (int v = tid; v < V_; v += 256) sum += expf(row[v] - m);
  red[tid] = sum; __syncthreads();
  for (int s = 128; s > 0; s >>= 1) { if (tid < s) red[tid] += red[tid + s]; __syncthreads(); }
  if (tid == 0) { rowmax[n] = m; rowlse[n] = logf(red[0]); }
}

// ---------------------------------------------------------------------------
// K4: logp = logits - max - lse  -> written straight into d_out step slot
// ---------------------------------------------------------------------------
__global__ void k_write_logp(const float* __restrict__ logits,
                             const float* __restrict__ rowmax,
                             const float* __restrict__ rowlse,
                             float* __restrict__ out) {
  size_t idx = (size_t)blockIdx.x * blockDim.x + threadIdx.x;
  int n = (int)(idx / V_);
  out[idx] = logits[idx] - rowmax[n] - rowlse[n];
}

// ---------------------------------------------------------------------------
// K5: top-8 over K*V per batch (block per batch, 512 threads)
// ---------------------------------------------------------------------------
__global__ void k_topk(const float* __restrict__ logits,
                       const float* __restrict__ rowmax,
                       const float* __restrict__ rowlse,
                       const float* __restrict__ ss_in,
                       float* __restrict__ step_ss,
                       int*   __restrict__ step_tok,
                       int*   __restrict__ step_pred,
                       float* __restrict__ ss_out,
                       int*   __restrict__ tok_out) {
  __shared__ float ls[512 * 8];
  __shared__ int   li[512 * 8];
  int b = blockIdx.x, tid = threadIdx.x;

  float base[K_];
  for (int k = 0; k < K_; ++k) {
    int n = b * K_ + k;
    base[k] = ss_in[n] - rowmax[n] - rowlse[n];   // ss + (logit - max - lse)
  }

  float bs[8]; int bi[8];
  for (int j = 0; j < 8; ++j) { bs[j] = -INFINITY; bi[j] = 0x7fffffff; }

  const int total = K_ * V_;
  for (int f = tid; f < total; f += 512) {
    int k = f / V_, v = f - k * V_;
    float val = base[k] + logits[(size_t)(b * K_ + k) * V_ + v];
    if (val > bs[7] || (val == bs[7] && f < bi[7])) {
      int j = 7;
      while (j > 0 && (val > bs[j-1] || (val == bs[j-1] && f < bi[j-1]))) {
        bs[j] = bs[j-1]; bi[j] = bi[j-1]; --j;
      }
      bs[j] = val; bi[j] = f;
    }
  }
  for (int j = 0; j < 8; ++j) { ls[tid * 8 + j] = bs[j]; li[tid * 8 + j] = bi[j]; }

  // tree merge of sorted-descending 8-lists
  for (int stride = 256; stride > 0; stride >>= 1) {
    __syncthreads();
    if (tid < stride) {
      float ms[8]; int mi[8];
      int p = 0, q = 0;
      for (int j = 0; j < 8; ++j) {
        float a = ls[tid * 8 + p];            int ai = li[tid * 8 + p];
        float c = ls[(tid + stride) * 8 + q]; int ci = li[(tid + stride) * 8 + q];
        bool takeA = (a > c) || (a == c && ai <= ci);
        if (takeA) { ms[j] = a; mi[j] = ai; ++p; }
        else       { ms[j] = c; mi[j] = ci; ++q; }
      }
      for (int j = 0; j < 8; ++j) { ls[tid * 8 + j] = ms[j]; li[tid * 8 + j] = mi[j]; }
    }
  }
  __syncthreads();
  if (tid == 0) {
    for (int j = 0; j < 8; ++j) {
      int n = b * K_ + j;
      int f = li[j];
      float sc = ls[j];
      int k = f / V_, v = f - k * V_;
      step_ss[n]   = sc;                     // unmasked score (used by final top-k)
      step_tok[n]  = v;
      step_pred[n] = b * K_ + k;             // global predecessor row
      tok_out[n]   = v;
      ss_out[n]    = (v == EOS_) ? NEGV : sc;
    }
  }
}

// ---------------------------------------------------------------------------
// K6: h <- h_new[pred]
// ---------------------------------------------------------------------------
__global__ void k_gather_h(const float* __restrict__ hn,
                           const int* __restrict__ pred,
                           float* __restrict__ h) {
  int n = blockIdx.x;
  int p = pred[n];
  for (int i = threadIdx.x; i < H_; i += blockDim.x)
    h[n * H_ + i] = hn[p * H_ + i];
}

// ---------------------------------------------------------------------------
// K7: backtrack (tiny; single thread) -> perm[t][n], sequences, score
// ---------------------------------------------------------------------------
__global__ void k_backtrack(const float* __restrict__ step_ss,
                            const int*   __restrict__ step_tok,
                            const int*   __restrict__ step_pred,
                            int*   __restrict__ perm,
                            float* __restrict__ out_seq,     // (B,K,T) as float
                            float* __restrict__ out_score) { // (B,K)
  if (threadIdx.x != 0 || blockIdx.x != 0) return;
  int tp[N_];
  for (int b = 0; b < B_; ++b) {
    float v[K_]; int id[K_];
    for (int k = 0; k < K_; ++k) { v[k] = step_ss[(T_ - 1) * N_ + b * K_ + k]; id[k] = k; }
    for (int i = 0; i < K_; ++i)
      for (int j = i + 1; j < K_; ++j)
        if (v[j] > v[i] || (v[j] == v[i] && id[j] < id[i])) {
          float tv = v[i]; v[i] = v[j]; v[j] = tv;
          int ti = id[i]; id[i] = id[j]; id[j] = ti;
        }
    for (int k = 0; k < K_; ++k) {
      out_score[b * K_ + k] = v[k];
      tp[b * K_ + k] = b * K_ + id[k];
    }
  }
  for (int t = T_ - 1; t >= 0; --t) {
    for (int n = 0; n < N_; ++n) {
      int p = tp[n];
      perm[t * N_ + n] = p;
      out_seq[n * T_ + t] = (float)step_tok[t * N_ + p];
    }
    for (int n = 0; n < N_; ++n) tp[n] = step_pred[t * N_ + tp[n]];
  }
}

// ---------------------------------------------------------------------------
// K8: in-place permute of the 8 beam rows per (t,b) in d_out, LDS-staged
// ---------------------------------------------------------------------------
__global__ void k_permute_out(float* __restrict__ out, const int* __restrict__ perm) {
  __shared__ float tile[8 * 1000];         // 32 KB
  int cx = blockIdx.x;                     // 0..31 (column chunk of 1000)
  int b  = blockIdx.y;                     // 0..7
  int t  = blockIdx.z;                     // 0..15
  int tid = threadIdx.x;
  size_t baseT = (size_t)t * N_ * V_;
  int c0 = cx * 1000;
  for (int idx = tid; idx < 8 * 1000; idx += blockDim.x) {
    int k = idx / 1000, c = idx - k * 1000;
    int src = perm[t * N_ + b * K_ + k];
    tile[idx] = out[baseT + (size_t)src * V_ + c0 + c];
  }
  __syncthreads();
  for (int idx = tid; idx < 8 * 1000; idx += blockDim.x) {
    int k = idx / 1000, c = idx - k * 1000;
    out[baseT + (size_t)(b * K_ + k) * V_ + c0 + c] = tile[idx];
  }
}

// ---------------------------------------------------------------------------
extern "C" void kernel_launch(void* const* d_in, const int* in_sizes, int n_in,
                              void* d_out, int out_size, void* d_ws, size_t ws_size,
                              hipStream_t stream) {
  const float* dec_hidden = (const float*)d_in[0];
  const float* q_enc      = (const float*)d_in[1];
  const int*   q_len      = (const int*)  d_in[2];
  const float* emb        = (const float*)d_in[3];
  const float* W_ih       = (const float*)d_in[4];
  const float* W_hh       = (const float*)d_in[5];
  const float* b_ih       = (const float*)d_in[6];
  const float* b_hh       = (const float*)d_in[7];
  const float* W_out      = (const float*)d_in[8];
  const float* b_out      = (const float*)d_in[9];
  float* out = (float*)d_out;

  // workspace layout (~10 MB)
  float* hbuf    = (float*)d_ws;                 // N*H
  float* hnew    = hbuf + N_ * H_;               // N*H
  float* xbuf    = hnew + N_ * H_;               // N*2H
  float* gibuf   = xbuf + N_ * 2 * H_;           // N*3H
  float* ghbuf   = gibuf + (size_t)N_ * G3H;     // N*3H
  float* logits  = ghbuf + (size_t)N_ * G3H;     // N*V
  float* rowmax  = logits + (size_t)N_ * V_;     // N
  float* rowlse  = rowmax + N_;                  // N
  float* ssbuf   = rowlse + N_;                  // N
  float* step_ss = ssbuf + N_;                   // T*N
  int* tokbuf    = (int*)(step_ss + T_ * N_);    // N
  int* step_tok  = tokbuf + N_;                  // T*N
  int* step_pred = step_tok + T_ * N_;           // T*N
  int* perm      = step_pred + T_ * N_;          // T*N

  const int LDS_A = N_ * KC_ * (int)sizeof(float);  // 128 KB (CDNA5 WGP has 320 KB)
  (void)hipFuncSetAttribute((const void*)k_gemm_wmma,
                            hipFuncAttributeMaxDynamicSharedMemorySize, LDS_A);

  k_init<<<N_, 256, 0, stream>>>(dec_hidden, hbuf, ssbuf, tokbuf);

  for (int t = 0; t < T_; ++t) {
    // X = [emb, ctx]
    k_attn_ctx<<<N_, 256, 0, stream>>>(hbuf, tokbuf, emb, q_enc, q_len, xbuf);
    // batched gate GEMMs (weights streamed once per step)
    k_gemm_wmma<<<G3H / 128, 256, LDS_A, stream>>>(xbuf, W_ih, b_ih, gibuf, 2 * H_, G3H);
    k_gemm_wmma<<<G3H / 128, 256, LDS_A, stream>>>(hbuf, W_hh, b_hh, ghbuf, H_, G3H);
    k_gru_elem<<<N_, 256, 0, stream>>>(gibuf, ghbuf, hbuf, hnew);
    // logits GEMM: W_out streamed exactly once (L2-resident, 65.5 MB)
    k_gemm_wmma<<<V_ / 128, 256, LDS_A, stream>>>(hnew, W_out, b_out, logits, H_, V_);
    k_row_stats<<<N_, 256, 0, stream>>>(logits, rowmax, rowlse);
    k_write_logp<<<(N_ * V_) / 256, 256, 0, stream>>>(logits, rowmax, rowlse,
                                                      out + (size_t)t * N_ * V_);
    k_topk<<<B_, 512, 0, stream>>>(logits, rowmax, rowlse, ssbuf,
                                   step_ss + t * N_, step_tok + t * N_, step_pred + t * N_,
                                   ssbuf, tokbuf);
    k_gather_h<<<N_, 256, 0, stream>>>(hnew, step_pred + t * N_, hbuf);
  }

  size_t TNV = (size_t)T_ * N_ * V_;
  k_backtrack<<<1, 32, 0, stream>>>(step_ss, step_tok, step_pred, perm,
                                    out + TNV, out + TNV + (size_t)N_ * T_);
  k_permute_out<<<dim3(32, B_, T_), 256, 0, stream>>>(out, perm);
}